// MultiHeadGATLayer_54769422959046
// MI455X (gfx1250) — compile-verified
//
#include <hip/hip_runtime.h>
#include <hip/hip_bf16.h>

typedef __attribute__((ext_vector_type(2))) float v2f;
typedef __attribute__((ext_vector_type(8))) float v8f;

#define NODES 2048
#define NB 4
#define HEADS 4
#define DHEAD 32
#define OUTF 128
#define INF_NEG (-3.402823e38f)
#define MASK_NEG (-1.0e9f)

// ---------------------------------------------------------------------------
// CDNA5 async global->LDS copy (ASYNCcnt path). The generic address of a
// __shared__ object has the LDS byte offset in its low 32 bits;
// GLOBAL_LOAD_ASYNC_TO_LDS_B128 takes the
// LDS byte address in VDST and the 64-bit global address in VADDR (GV mode).
// ---------------------------------------------------------------------------
__device__ __forceinline__ void async_load_b128(const float* g, float* l) {
  const unsigned lds = (unsigned)(uintptr_t)l;  // LDS byte offset
  const unsigned long long ga = (unsigned long long)(uintptr_t)g;
  asm volatile("global_load_async_to_lds_b128 %0, %1, off"
               :: "v"(lds), "v"(ga) : "memory");
}
__device__ __forceinline__ void async_wait0() {
  asm volatile("s_wait_asynccnt 0x0" ::: "memory");
}

// ---------------------------------------------------------------------------
// Kernel 1: h = x @ W   (M = B*N = 8192, K = 64, N = 128), fp32 WMMA 16x16x4.
// Block = 256 thr = 8 waves. Block owns 16 rows; wave w owns 16-col tile.
// A tile staged via async-to-LDS; stride 68 floats (272B) keeps every b128
// destination 16B-aligned and reads bank-conflict-free (bank = 4m+k mod 64).
// ---------------------------------------------------------------------------
__global__ void gemm_h_kernel(const float* __restrict__ x,
                              const float* __restrict__ W,
                              float* __restrict__ hbuf) {
  __shared__ __align__(16) float As[16 * 68];
  const int lane = threadIdx.x & 31;
  const int wid  = threadIdx.x >> 5;
  const int row0 = blockIdx.x * 16;

  // Stage 16x64 A tile: each thread issues one async b128 (no VGPR roundtrip).
  {
    const int r  = threadIdx.x >> 4;  // 0..15
    const int c4 = threadIdx.x & 15;  // 0..15 -> cols c4*4..c4*4+3
    async_load_b128(x + (size_t)(row0 + r) * 64 + c4 * 4, &As[r * 68 + c4 * 4]);
  }
  async_wait0();
  __syncthreads();

  const int n0   = wid * 16;
  const int nn   = lane & 15;
  const int half = lane >> 4;
  const int m    = lane & 15;

  v8f c = {};
#pragma unroll
  for (int k0 = 0; k0 < 64; k0 += 4) {
    const int k = k0 + 2 * half;
    v2f a, b;
    // A 16x4 f32 layout: lane m, VGPR0 = K(2*half), VGPR1 = K(2*half+1)
    a.x = As[m * 68 + k];
    a.y = As[m * 68 + k + 1];
    // B 4x16 f32 layout (row striped across lanes, K across VGPR/half)
    b.x = W[(size_t)k * OUTF + n0 + nn];
    b.y = W[(size_t)(k + 1) * OUTF + n0 + nn];
    c = __builtin_amdgcn_wmma_f32_16x16x4_f32(false, a, false, b,
                                              (short)0, c, false, false);
  }

  // D layout: VGPR r -> M = r + 8*half, N = lane&15
#pragma unroll
  for (int r = 0; r < 8; ++r) {
    const int mr = r + 8 * half;
    hbuf[(size_t)(row0 + mr) * OUTF + n0 + nn] = c[r];
  }
}

// ---------------------------------------------------------------------------
// Kernel 2: s_src[b,n,h] = sum_d h*a_src ; s_dst likewise. Wave per row.
// ---------------------------------------------------------------------------
__global__ void scores_kernel(const float* __restrict__ hbuf,
                              const float* __restrict__ a_src,
                              const float* __restrict__ a_dst,
                              float* __restrict__ ssrc,
                              float* __restrict__ sdst) {
  const int lane = threadIdx.x & 31;
  const int wid  = threadIdx.x >> 5;
  const int row  = blockIdx.x * 8 + wid;  // 0..8191

#pragma unroll
  for (int h = 0; h < HEADS; ++h) {
    const float hv = hbuf[(size_t)row * OUTF + h * DHEAD + lane];
    float vs = hv * a_src[h * DHEAD + lane];
    float vd = hv * a_dst[h * DHEAD + lane];
#pragma unroll
    for (int off = 16; off > 0; off >>= 1) {
      vs += __shfl_xor(vs, off);
      vd += __shfl_xor(vd, off);
    }
    if (lane == 0) {
      ssrc[(size_t)row * HEADS + h] = vs;
      sdst[(size_t)row * HEADS + h] = vd;
    }
  }
}

// ---------------------------------------------------------------------------
// Kernel 3: one wave per (b,i). Streaming top-8 per head, exact softmax over
// the 8 survivors (dropped entries underflow to exactly 0 in fp32), sparse
// gather for out, LDS scatter + dense row write for alpha_mean.
// Block = 128 thr = 4 waves; LDS = 4 * (2048 mask + 2048 alpha) f32 = 64 KB.
// Mask row staged via async-to-LDS (16 x b128 per wave).
// ---------------------------------------------------------------------------
__global__ void gat_attend_kernel(const float* __restrict__ hbuf,
                                  const float* __restrict__ ssrc,
                                  const float* __restrict__ sdst,
                                  const float* __restrict__ mask,
                                  float* __restrict__ out,
                                  float* __restrict__ amean) {
  extern __shared__ __align__(16) float smem[];
  const int lane = threadIdx.x & 31;
  const int wid  = threadIdx.x >> 5;
  const int row  = blockIdx.x * 4 + wid;  // b*N + i
  const int b    = row >> 11;
  const int i    = row & (NODES - 1);

  float* mask_s  = smem + wid * 4096;
  float* alpha_s = mask_s + NODES;

  // Phase 1: async-stage mask row into LDS, zero alpha row.
  const float* mrow = mask + (size_t)i * NODES;
  __builtin_prefetch(mask + (size_t)((i + 4) & (NODES - 1)) * NODES, 0, 1);
#pragma unroll
  for (int t = 0; t < 16; ++t) {
    const int j4 = t * 32 + lane;
    async_load_b128(mrow + j4 * 4, mask_s + j4 * 4);
    ((float4*)alpha_s)[j4] = make_float4(0.f, 0.f, 0.f, 0.f);
  }
  async_wait0();
  __syncthreads();

  const float* sd_b = sdst + (size_t)b * NODES * HEADS;
  int   sa_idx[HEADS];
  float sa_val[HEADS];
#pragma unroll
  for (int h = 0; h < HEADS; ++h) { sa_idx[h] = 0; sa_val[h] = 0.f; }

#pragma unroll
  for (int h = 0; h < HEADS; ++h) {
    const float si = ssrc[(size_t)row * HEADS + h];

    // Per-lane streaming top-8 (64 candidates per lane).
    float tv[8]; int ti[8];
#pragma unroll
    for (int k = 0; k < 8; ++k) { tv[k] = INF_NEG; ti[k] = 0; }

    for (int t = 0; t < 64; ++t) {
      const int j = t * 32 + lane;
      const float pre = si + sd_b[j * HEADS + h];
      float e = (pre >= 0.f) ? pre : 0.2f * pre;      // leaky_relu 0.2
      const float m = mask_s[j];
      e = e * m + (1.f - m) * MASK_NEG;               // soft mask blend
      if (e > tv[7]) {
        tv[7] = e; ti[7] = j;
#pragma unroll
        for (int k = 7; k > 0; --k)
          if (tv[k] > tv[k - 1]) {
            float fv = tv[k]; tv[k] = tv[k - 1]; tv[k - 1] = fv;
            int   fi = ti[k]; ti[k] = ti[k - 1]; ti[k - 1] = fi;
          }
      }
    }

    // Wave-wide merge: 8 argmax pops over lanes' sorted heads.
    float wv[8]; int wi[8];
#pragma unroll
    for (int k = 0; k < 8; ++k) {
      float bv = tv[0]; int bi = ti[0]; int bl = lane;
#pragma unroll
      for (int off = 16; off > 0; off >>= 1) {
        const float ov = __shfl_xor(bv, off);
        const int   oi = __shfl_xor(bi, off);
        const int   ol = __shfl_xor(bl, off);
        if (ov > bv || (ov == bv && ol < bl)) { bv = ov; bi = oi; bl = ol; }
      }
      wv[k] = bv; wi[k] = bi;
      if (lane == bl) {  // winner pops its head
#pragma unroll
        for (int k2 = 0; k2 < 7; ++k2) { tv[k2] = tv[k2 + 1]; ti[k2] = ti[k2 + 1]; }
        tv[7] = INF_NEG;
      }
    }

    // Softmax over the 8 kept (wv[0] is the global max).
    const float mx = wv[0];
    float p[8]; float denom = 0.f;
#pragma unroll
    for (int k = 0; k < 8; ++k) { p[k] = __expf(wv[k] - mx); denom += p[k]; }
    const float inv = 1.f / denom;

    // Sparse aggregation: lane = d in [0,32).
    float acc = 0.f;
#pragma unroll
    for (int k = 0; k < 8; ++k) {
      const float a = p[k] * inv;
      acc += a * hbuf[(size_t)(b * NODES + wi[k]) * OUTF + h * DHEAD + lane];
      if (lane == k) { sa_val[h] = a; sa_idx[h] = wi[k]; }
    }
    out[(size_t)row * OUTF + h * DHEAD + lane] = acc;
  }
  __syncthreads();

  // Scatter alpha/H into the zeroed LDS row (lanes 0..7 hold the winners).
#pragma unroll
  for (int h = 0; h < HEADS; ++h)
    if (lane < 8) atomicAdd(&alpha_s[sa_idx[h]], 0.25f * sa_val[h]);
  __syncthreads();

  // Dense write of alpha_mean row.
  float* arow = amean + (size_t)row * NODES;
#pragma unroll
  for (int t = 0; t < 16; ++t)
    ((float4*)arow)[t * 32 + lane] = ((float4*)alpha_s)[t * 32 + lane];
}

// ---------------------------------------------------------------------------
extern "C" void kernel_launch(void* const* d_in, const int* in_sizes, int n_in,
                              void* d_out, int out_size, void* d_ws, size_t ws_size,
                              hipStream_t stream) {
  const float* x     = (const float*)d_in[0];  // (4,2048,64)
  const float* W     = (const float*)d_in[1];  // (64,128)
  const float* a_src = (const float*)d_in[2];  // (4,32)
  const float* a_dst = (const float*)d_in[3];  // (4,32)
  const float* emask = (const float*)d_in[4];  // (2048,2048)

  float* out_feat = (float*)d_out;                         // 4*2048*128
  float* out_am   = out_feat + (size_t)NB * NODES * OUTF;  // 4*2048*2048

  float* hbuf = (float*)d_ws;                        // 8192*128 f32 (4 MB)
  float* ssrc = hbuf + (size_t)NB * NODES * OUTF;    // 8192*4
  float* sdst = ssrc + (size_t)NB * NODES * HEADS;   // 8192*4

  gemm_h_kernel<<<(NB * NODES) / 16, 256, 0, stream>>>(x, W, hbuf);
  scores_kernel<<<(NB * NODES) / 8, 256, 0, stream>>>(hbuf, a_src, a_dst, ssrc, sdst);
  gat_attend_kernel<<<(NB * NODES) / 4, 128, 4 * 4096 * sizeof(float), stream>>>(
      hbuf, ssrc, sdst, emask, out_feat, out_am);
}